// GraphConv_352187318910
// MI455X (gfx1250) — compile-verified
//
#include <hip/hip_runtime.h>
#include <hip/hip_bf16.h>
#include <stdint.h>

#define D_DIM 256
#define BN_EPS 1e-5f

typedef __attribute__((ext_vector_type(2))) float v2f;
typedef __attribute__((ext_vector_type(8))) float v8f;

// ---------------------------------------------------------------------------
// Kernel 1: zero the output accumulator, init degree to 1.0 (self loop)
// ---------------------------------------------------------------------------
__global__ __launch_bounds__(256) void gcn_init(float* __restrict__ out,
                                                float* __restrict__ deg,
                                                int n_nodes, long long n_elems) {
    long long idx = (long long)blockIdx.x * blockDim.x + threadIdx.x;
    if (idx < n_elems) out[idx] = 0.0f;
    if (idx < n_nodes) deg[idx] = 1.0f;   // self-loop contributes 1 to in-degree
}

// ---------------------------------------------------------------------------
// Kernel 2: in-degree histogram over dst
// ---------------------------------------------------------------------------
__global__ __launch_bounds__(256) void gcn_degree(const long long* __restrict__ dst,
                                                  float* __restrict__ deg, int E) {
    int e = blockIdx.x * blockDim.x + threadIdx.x;
    if (e < E) {
        unsafeAtomicAdd(&deg[(int)dst[e]], 1.0f);
    }
}

// ---------------------------------------------------------------------------
// Kernel 3: dinv = deg > 0 ? rsqrt(deg) : 0
// ---------------------------------------------------------------------------
__global__ __launch_bounds__(256) void gcn_dinv(const float* __restrict__ deg,
                                                float* __restrict__ dinv, int n) {
    int i = blockIdx.x * blockDim.x + threadIdx.x;
    if (i < n) {
        float d = deg[i];
        dinv[i] = (d > 0.0f) ? rsqrtf(d) : 0.0f;
    }
}

// ---------------------------------------------------------------------------
// Kernel 4: H = X @ W with v_wmma_f32_16x16x4_f32.
// One wave computes a 16-row x 64-col strip (4 WMMA tiles). A block of 8
// waves shares a double-buffered 64x64 LDS slice of W staged with
// GLOBAL_LOAD_ASYNC_TO_LDS_B128 (ASYNCcnt), so the copy of K-slice kt+1
// overlaps the WMMAs of K-slice kt. Stride 65 avoids bank conflicts.
// Requires n_rows % 16 == 0 (50000 = 3125*16).
//
// f32 WMMA operand layout (ISA 7.12.2, 32-bit A 16x4):
//   lane = 16*half + r;  a[v] = X[m0+r][k + v + 2*half]   (contiguous pair)
//   b[v] = W[k + v + 2*half][n0 + r]
//   C/D:  acc[v] = H[m0 + v + 8*half][n0 + r]
// ---------------------------------------------------------------------------
__global__ __launch_bounds__(256) void gcn_gemm_wmma(const float* __restrict__ X,
                                                     const float* __restrict__ Wm,
                                                     float* __restrict__ H,
                                                     int nRowTiles) {
    const int lane = threadIdx.x & 31;
    const int wave = threadIdx.x >> 5;
    const int half = lane >> 4;   // 0..1
    const int r    = lane & 15;   // 0..15
    const int rowTile = blockIdx.x * 8 + wave;
    const int n0 = blockIdx.y * 64;           // 4 tiles of 16 columns
    const bool active = (rowTile < nRowTiles);
    const int m0 = rowTile * 16;

    __shared__ float lw[2][64 * 65];          // double-buffered W slice

    v8f acc0 = {}, acc1 = {}, acc2 = {}, acc3 = {};

    // Issue async global->LDS copies of W[k0:k0+64, n0:n0+64] into buffer buf.
    // 1024 float4 total; each of the 256 threads issues 4 b128 copies.
    auto stage = [&](int k0, int buf) {
        #pragma unroll
        for (int i = 0; i < 4; ++i) {
            int f4 = threadIdx.x + i * 256;       // 0..1023
            int kk = f4 >> 4;                     // K-row within slice
            int c4 = (f4 & 15) << 2;              // col*4
            // LDS byte address = low 32 bits of the flat shared-aperture ptr
            uint32_t ldsa = (uint32_t)(uintptr_t)&lw[buf][kk * 65 + c4];
            const float* gp = Wm + (size_t)(k0 + kk) * D_DIM + n0 + c4;
            asm volatile("global_load_async_to_lds_b128 %0, %1, off"
                         :: "v"(ldsa), "v"(gp) : "memory");
        }
    };

    stage(0, 0);                              // prologue copy of slice 0

    for (int kt = 0; kt < D_DIM / 64; ++kt) {
        const int buf = kt & 1;
        // Wait for this wave's outstanding async copies, then make all
        // waves' LDS writes visible to the whole workgroup.
        asm volatile("s_wait_asynccnt 0x0" ::: "memory");
        __syncthreads();
        if (kt + 1 < D_DIM / 64) stage((kt + 1) * 64, buf ^ 1);   // overlap

        if (active) {
            const int k0 = kt * 64;
            const float* xrow = X + (size_t)(m0 + r) * D_DIM + k0;
            const float* lb   = &lw[buf][0];
            #pragma unroll 4
            for (int kk = 0; kk < 64; kk += 4) {
                const int kg = kk + 2 * half;
                v2f a = *(const v2f*)(xrow + kg);
                const float* lb0 = lb + kg * 65;
                const float* lb1 = lb + (kg + 1) * 65;
                v2f b0 = { lb0[r +  0], lb1[r +  0] };
                v2f b1 = { lb0[r + 16], lb1[r + 16] };
                v2f b2 = { lb0[r + 32], lb1[r + 32] };
                v2f b3 = { lb0[r + 48], lb1[r + 48] };
                acc0 = __builtin_amdgcn_wmma_f32_16x16x4_f32(false, a, false, b0, (short)0, acc0, false, false);
                acc1 = __builtin_amdgcn_wmma_f32_16x16x4_f32(false, a, false, b1, (short)0, acc1, false, false);
                acc2 = __builtin_amdgcn_wmma_f32_16x16x4_f32(false, a, false, b2, (short)0, acc2, false, false);
                acc3 = __builtin_amdgcn_wmma_f32_16x16x4_f32(false, a, false, b3, (short)0, acc3, false, false);
            }
        }
        __syncthreads();   // protect buf^1 from being restaged while read
    }

    if (active) {
        #pragma unroll
        for (int v = 0; v < 8; ++v) {
            size_t row = (size_t)(m0 + v + 8 * half) * D_DIM;
            H[row + n0 +  0 + r] = acc0[v];
            H[row + n0 + 16 + r] = acc1[v];
            H[row + n0 + 32 + r] = acc2[v];
            H[row + n0 + 48 + r] = acc3[v];
        }
    }
}

// ---------------------------------------------------------------------------
// Kernel 5: edge scatter. One wave per edge; each lane moves 2 float4 of
// h[src], scales by dinv[src]*dinv[dst], atomically adds to out[dst].
// ---------------------------------------------------------------------------
__global__ __launch_bounds__(256) void gcn_scatter(const long long* __restrict__ srcI,
                                                   const long long* __restrict__ dstI,
                                                   const float* __restrict__ H,
                                                   const float* __restrict__ dinv,
                                                   float* __restrict__ out, int E) {
    int edge = blockIdx.x * 8 + (threadIdx.x >> 5);
    if (edge >= E) return;
    const int lane = threadIdx.x & 31;
    const int s = (int)srcI[edge];
    const int d = (int)dstI[edge];
    const float norm = dinv[s] * dinv[d];
    const float4* hp = (const float4*)(H + (size_t)s * D_DIM);
    float* op = out + (size_t)d * D_DIM;
    __builtin_prefetch(hp + lane, 0, 0);      // global_prefetch_b8
    #pragma unroll
    for (int g = 0; g < 2; ++g) {
        int q = lane + g * 32;                // float4 index within the row
        float4 v = hp[q];
        int c = q * 4;
        unsafeAtomicAdd(op + c + 0, v.x * norm);
        unsafeAtomicAdd(op + c + 1, v.y * norm);
        unsafeAtomicAdd(op + c + 2, v.z * norm);
        unsafeAtomicAdd(op + c + 3, v.w * norm);
    }
}

// ---------------------------------------------------------------------------
// Kernel 6: self-loop term + bias + BatchNorm(eval) + ReLU, in place.
// ---------------------------------------------------------------------------
__global__ __launch_bounds__(256) void gcn_finalize(float* __restrict__ out,
                                                    const float* __restrict__ H,
                                                    const float* __restrict__ dinv,
                                                    const float* __restrict__ bias,
                                                    const float* __restrict__ gamma,
                                                    const float* __restrict__ beta,
                                                    const float* __restrict__ rmean,
                                                    const float* __restrict__ rvar,
                                                    long long n_elems) {
    long long idx = (long long)blockIdx.x * blockDim.x + threadIdx.x;
    if (idx >= n_elems) return;
    int i = (int)(idx >> 8);   // node  (D_DIM == 256)
    int c = (int)(idx & 255);  // channel
    float di = dinv[i];
    float val = out[idx] + H[idx] * di * di + bias[c];
    float y = (val - rmean[c]) * rsqrtf(rvar[c] + BN_EPS) * gamma[c] + beta[c];
    out[idx] = fmaxf(y, 0.0f);
}

// ---------------------------------------------------------------------------
extern "C" void kernel_launch(void* const* d_in, const int* in_sizes, int n_in,
                              void* d_out, int out_size, void* d_ws, size_t ws_size,
                              hipStream_t stream) {
    const float*     x     = (const float*)d_in[0];
    const long long* ei    = (const long long*)d_in[1];   // [2, E] row-major
    const float*     W     = (const float*)d_in[2];
    const float*     b     = (const float*)d_in[3];
    const float*     gamma = (const float*)d_in[4];
    const float*     beta  = (const float*)d_in[5];
    const float*     rmean = (const float*)d_in[6];
    const float*     rvar  = (const float*)d_in[7];
    float* out = (float*)d_out;

    const int N = in_sizes[0] / D_DIM;        // 50000
    const int E = in_sizes[1] / 2;            // 800000
    const long long NE = (long long)N * D_DIM;

    const long long* srcI = ei;               // edge_index[0]
    const long long* dstI = ei + E;           // edge_index[1]

    // Workspace: h [N*256] f32, deg [N] f32, dinv [N] f32
    float* h    = (float*)d_ws;
    float* deg  = h + NE;
    float* dinv = deg + N;

    // 1) zero out + deg=1
    {
        long long blocks = (NE + 255) / 256;
        gcn_init<<<(int)blocks, 256, 0, stream>>>(out, deg, N, NE);
    }
    // 2) in-degree
    gcn_degree<<<(E + 255) / 256, 256, 0, stream>>>(dstI, deg, E);
    // 3) dinv
    gcn_dinv<<<(N + 255) / 256, 256, 0, stream>>>(deg, dinv, N);
    // 4) H = X @ W  (N is a multiple of 16)
    {
        int nRowTiles = N / 16;               // 3125
        dim3 grid((nRowTiles + 7) / 8, D_DIM / 64);
        gcn_gemm_wmma<<<grid, 256, 0, stream>>>(x, W, h, nRowTiles);
    }
    // 5) edge scatter-add
    gcn_scatter<<<(E + 7) / 8, 256, 0, stream>>>(srcI, dstI, h, dinv, out, E);
    // 6) self-loop + bias + BN + ReLU
    {
        long long blocks = (NE + 255) / 256;
        gcn_finalize<<<(int)blocks, 256, 0, stream>>>(out, h, dinv, b, gamma, beta,
                                                      rmean, rvar, NE);
    }
}